// Postprocess_19928648253795
// MI455X (gfx1250) — compile-verified
//
#include <hip/hip_runtime.h>
#include <stdint.h>

// ---------------------------------------------------------------------------
// SMLM-style 3D peak post-processing for MI455X (gfx1250).
// Separable 5x5x5 stencil (max-pool NMS + box-sum + x-moment), streamed in z
// with CDNA5 async global->LDS double buffering (ASYNCcnt pipeline).
// Roofline: 100 MB read + 503 MB write => ~26 us floor at 23.3 TB/s; the
// separable LDS scheme (~45 VALU ops/voxel) keeps the kernel bandwidth-bound.
// Whole input (100 MB) is resident in the 192 MB L2, so halo re-reads are L2 hits.
// ---------------------------------------------------------------------------

#define D_ 96
#define H_ 512
#define W_ 512
#define TX 64          // output tile width  (x)
#define TY 8           // output tile height (y)
#define ZCHUNK 24      // z planes per block (D_/gridDim.z)
#define THRESH 0.3f

#ifndef __has_builtin
#define __has_builtin(x) 0
#endif

#if __has_builtin(__builtin_amdgcn_global_load_async_to_lds_b32)
#define USE_ASYNC_BUILTIN 1
#else
#define USE_ASYNC_BUILTIN 0
#endif
#if __has_builtin(__builtin_amdgcn_s_wait_asynccnt)
#define USE_WAITASYNC_BUILTIN 1
#else
#define USE_WAITASYNC_BUILTIN 0
#endif

typedef __attribute__((address_space(1))) int* gas_ip;   // global (as1) int*
typedef __attribute__((address_space(3))) int* las_ip;   // LDS    (as3) int*

// One fp32 element: global memory -> LDS, tracked on ASYNCcnt (no VGPR data).
__device__ __forceinline__ void async_ld_f32(const float* g, float* l) {
#if USE_ASYNC_BUILTIN
    __builtin_amdgcn_global_load_async_to_lds_b32(
        (gas_ip)(uintptr_t)g,
        (las_ip)(unsigned int)(uintptr_t)l,   // low 32 bits of flat addr = LDS offset
        0, 0);
#else
    asm volatile("global_load_async_to_lds_b32 %0, %1, off"
                 :
                 : "v"((unsigned int)(uintptr_t)l), "v"(g)
                 : "memory");
#endif
}

__device__ __forceinline__ void wait_async0() {
#if USE_WAITASYNC_BUILTIN
    __builtin_amdgcn_s_wait_asynccnt(0);
#else
    asm volatile("s_wait_asynccnt 0" ::: "memory");
#endif
}

__global__ __launch_bounds__(256)
void Postprocess_19928648253795_kernel(const float* __restrict__ pred,
                                       float* __restrict__ out) {
    // ---- LDS (~55 KB of the 320 KB WGP pool) -----------------------------
    __shared__ float s_raw [2][TY + 4][TX + 4];   // double-buffered z-plane tile (+halo)
    __shared__ float s_hmax[TY + 4][TX];          // x-pass results (thresholded max)
    __shared__ float s_hsum[TY + 4][TX];          // x-pass box sum
    __shared__ float s_hsx [TY + 4][TX];          // x-pass moment sum
    __shared__ float s_zmax[5][TY][TX];           // per-plane (x,y reduced), 5-deep ring
    __shared__ float s_zsum[5][TY][TX];
    __shared__ float s_zsx [5][TY][TX];
    __shared__ float s_ctr [5][TY][TX];           // raw center values, 5-deep ring

    const int tid = threadIdx.x;
    const int x0  = blockIdx.x * TX;
    const int y0  = blockIdx.y * TY;
    const int zs  = blockIdx.z * ZCHUNK;
    const int ze  = zs + ZCHUNK;
    const size_t N = (size_t)D_ * H_ * W_;

    // ---- stage plane z into s_raw[b] via async global->LDS ---------------
    auto issue_load = [&](int b, int z) {
        for (int i = tid; i < (TY + 4) * (TX + 4); i += 256) {
            const int yy = i / (TX + 4);
            const int xx = i - yy * (TX + 4);
            const int gy = y0 + yy - 2;
            const int gx = x0 + xx - 2;
            float* ldst = &s_raw[b][yy][xx];
            if ((unsigned)gy < (unsigned)H_ && (unsigned)gx < (unsigned)W_) {
                const float* gsrc = pred + ((size_t)z * H_ + gy) * W_ + gx;
                async_ld_f32(gsrc, ldst);           // EXEC-masked async copy
            } else {
                *ldst = 0.0f;                       // zero padding (matches conv pad)
            }
        }
    };

    // ---- horizontal 5-tap pass over (TY+4) rows --------------------------
    auto hpass = [&](int b) {
        for (int i = tid; i < (TY + 4) * TX; i += 256) {
            const int yy = i >> 6;            // TX == 64
            const int xx = i & (TX - 1);
            const float* r = &s_raw[b][yy][xx];
            const float a0 = r[0], a1 = r[1], a2 = r[2], a3 = r[3], a4 = r[4];
            s_hsum[yy][xx] = a0 + a1 + a2 + a3 + a4;
            // x-moment weights from fv = [-1,-2,0,2,1] (cross-correlation)
            s_hsx [yy][xx] = -a0 - 2.0f * a1 + 2.0f * a3 + a4;
            const float t0 = a0 > THRESH ? a0 : 0.0f;
            const float t1 = a1 > THRESH ? a1 : 0.0f;
            const float t2 = a2 > THRESH ? a2 : 0.0f;
            const float t3 = a3 > THRESH ? a3 : 0.0f;
            const float t4 = a4 > THRESH ? a4 : 0.0f;
            s_hmax[yy][xx] = fmaxf(fmaxf(fmaxf(t0, t1), fmaxf(t2, t3)), t4);
        }
    };

    // ---- vertical 5-tap pass -> ring buffer slot z%5 ---------------------
    auto vpass = [&](int b, int z) {
        const int s = z % 5;
        for (int i = tid; i < TY * TX; i += 256) {
            const int y = i >> 6;
            const int x = i & (TX - 1);
            float m  = s_hmax[y][x];
            float S  = s_hsum[y][x];
            float Sx = s_hsx [y][x];
#pragma unroll
            for (int j = 1; j < 5; ++j) {
                m  = fmaxf(m, s_hmax[y + j][x]);
                S  += s_hsum[y + j][x];
                Sx += s_hsx [y + j][x];
            }
            s_zmax[s][y][x] = m;
            s_zsum[s][y][x] = S;
            s_zsx [s][y][x] = Sx;
            s_ctr [s][y][x] = s_raw[b][y + 2][x + 2];
        }
    };

    // ---- combine 5 buffered planes, emit outputs for plane zo ------------
    auto emit = [&](int zo) {
        const int lo = (zo - 2 < 0) ? 0 : zo - 2;           // missing planes pad with
        const int hi = (zo + 2 > D_ - 1) ? D_ - 1 : zo + 2; // 0-sum / no-op max (pt>=0)
        for (int i = tid; i < TY * TX; i += 256) {
            const int y = i >> 6;
            const int x = i & (TX - 1);
            float mp = 0.0f, S = 0.0f, Sx = 0.0f;
            for (int zp = lo; zp <= hi; ++zp) {
                const int s = zp % 5;
                mp = fmaxf(mp, s_zmax[s][y][x]);
                S  += s_zsum[s][y][x];
                Sx += s_zsx [s][y][x];
            }
            const float c    = s_ctr[zo % 5][y][x];
            const float pt   = c > THRESH ? c : 0.0f;
            const float conf = (mp > 0.0f && mp == pt) ? mp : 0.0f;  // exact: maxes copy bits
            const float s0   = (S != 0.0f) ? S : 1.0f;
            const float xloc = Sx / s0;
            const int gx = x0 + x, gy = y0 + y;
            const size_t idx = ((size_t)zo * H_ + gy) * W_ + gx;
            // Reference bug replicated: y/z recoveries also use xloc.
            const float xr = ((float)gx + xloc - 256.0f + 0.5f) * 0.25f;
            const float yr = ((float)gy + xloc - 256.0f + 0.5f) * 0.25f;
            const float zr = ((float)zo + xloc + 0.5f) * 0.1f - 2.0f;
            const float m  = conf > 0.0f ? 1.0f : 0.0f;
            out[idx * 3 + 0] = xr * m;          // xyz  [D,H,W,3]
            out[idx * 3 + 1] = yr * m;
            out[idx * 3 + 2] = zr * m;
            out[3 * N + idx] = conf;            // conf [D,H,W]
            out[4 * N + idx] = m;               // mask [D,H,W] as 0/1
        }
    };

    auto loadable = [&](int z) { return z >= 0 && z < D_; };

    // ---- z-streamed, async-double-buffered pipeline ----------------------
    const int zfirst = zs - 2;
    if (loadable(zfirst)) issue_load(zfirst & 1, zfirst);
    wait_async0();
    __syncthreads();

    for (int zi = zfirst; zi <= ze + 1; ++zi) {
        const int b  = zi & 1;
        const int zn = zi + 1;
        if (zn <= ze + 1 && loadable(zn)) issue_load(zn & 1, zn); // prefetch next plane
        if (loadable(zi)) {
            hpass(b);
            __syncthreads();          // h-arrays visible across waves
            vpass(b, zi);
        }
        __syncthreads();              // ring-buffer slot zi%5 visible
        const int zo = zi - 2;
        if (zo >= zs) emit(zo);
        wait_async0();                // next plane landed in LDS
        __syncthreads();              // everyone done with raw[b] / h-arrays / old slots
    }
}

extern "C" void kernel_launch(void* const* d_in, const int* in_sizes, int n_in,
                              void* d_out, int out_size, void* d_ws, size_t ws_size,
                              hipStream_t stream) {
    const float* pred = (const float*)d_in[0];   // pred_vol [1,1,96,512,512] fp32
    // d_in[1] (local_filter) is fully determined by R=2; weights are hardcoded.
    float* out = (float*)d_out;                  // xyz | conf | mask, concatenated
    dim3 grid(W_ / TX, H_ / TY, D_ / ZCHUNK);    // (8, 64, 4)
    Postprocess_19928648253795_kernel<<<grid, dim3(256), 0, stream>>>(pred, out);
}